// Model_11888469475824
// MI455X (gfx1250) — compile-verified
//
#include <hip/hip_runtime.h>

typedef __attribute__((ext_vector_type(2))) float v2f;
typedef __attribute__((ext_vector_type(8))) float v8f;

#define IN_CH   12
#define OUT_CH  16
#define KS      5
#define H       384
#define W       384
#define OH      192
#define OW      192
#define HW      (H * W)       // 147456
#define OHW     (OH * OW)     // 36864
#define KTOT    (IN_CH * KS * KS) // 300
#define BATCH   32
#define XTOT    (BATCH * IN_CH * HW)  // 56623104 floats (fits int32)

// Workspace layout (floats)
#define WS_INSUM    0     // 384:  sum of x over spatial, per (b, in_ch)
#define WS_CONVSUM  384   // 512:  sum of hardswish(conv) per (b, out_ch)
#define WS_SCALE    896   // 512:  1 + gating per (b, out_ch)
#define WS_ADD      1408  // 512:  residual projection per (b, out_ch)
#define WS_TOTAL    1920

// ---------------------------------------------------------------------------
// Kernel 1: per-(b, in_ch) global sum of x. One block per channel plane.
// ---------------------------------------------------------------------------
__global__ __launch_bounds__(256) void pool_input_kernel(
    const float* __restrict__ x, float* __restrict__ ws)
{
    int ch = blockIdx.x;  // b * IN_CH + c
    const float4* p = (const float4*)(x + (size_t)ch * HW);
    float s = 0.0f;
    for (int i = threadIdx.x; i < HW / 4; i += 256) {
        float4 v = p[i];
        s += (v.x + v.y) + (v.z + v.w);
    }
    __shared__ float red[256];
    red[threadIdx.x] = s;
    __syncthreads();
    for (int off = 128; off > 0; off >>= 1) {
        if (threadIdx.x < off) red[threadIdx.x] += red[threadIdx.x + off];
        __syncthreads();
    }
    if (threadIdx.x == 0) ws[WS_INSUM + ch] = red[0];
}

// ---------------------------------------------------------------------------
// Templated K-loop for the implicit-GEMM conv.
//   CJ:   the single N-fragment (j) that needs x-bounds handling (-1 = none)
//   YCHK: tile touches the top/bottom padded rows (oh==0 or oh==OH-1);
//         then every fragment is fully checked (rare: ~1% of tiles).
// Checked loads are branch-free: clamp flat index into the x allocation,
// load unconditionally, select 0 via cndmask.
// ---------------------------------------------------------------------------
template<int CJ, bool YCHK>
__device__ __forceinline__ void conv_loop(
    const float* __restrict__ x, const float* __restrict__ wlds,
    const int2* __restrict__ kdec, int nl, int lh,
    int oh2, int posabs, int ixbase, v8f acc[4])
{
    for (int kb = 0; kb < KTOT; kb += 4) {
        int k0 = kb + lh * 2;
        v2f a;
        a.x = wlds[nl * KTOT + k0];
        a.y = wlds[nl * KTOT + k0 + 1];

        int2 d0 = kdec[k0];
        int2 d1 = kdec[k0 + 1];
        int ib0 = posabs + d0.x;            // absolute flat index, j=0 lane base
        int ib1 = posabs + d1.x;

        int kxm0 = 0, kxm1 = 0, ixq0 = 0, ixq1 = 0;
        bool y0 = true, y1 = true;
        if (YCHK || CJ >= 0) {
            kxm0 = (d0.y << 24) >> 24;      // sign-extended (kx-2)
            kxm1 = (d1.y << 24) >> 24;
            ixq0 = ixbase + kxm0;           // ix for j=0 (add 32*j per fragment)
            ixq1 = ixbase + kxm1;
        }
        if (YCHK) {
            int ky0 = d0.y >> 8, ky1 = d1.y >> 8;
            y0 = (unsigned)(oh2 + ky0 - 2) < H;
            y1 = (unsigned)(oh2 + ky1 - 2) < H;
        }

#pragma unroll
        for (int j = 0; j < 4; j++) {
            int o = 32 * j;
            v2f bf;
            if (YCHK || j == CJ) {
                int i0 = ib0 + o, i1 = ib1 + o;
                int c0 = i0 < 0 ? 0 : (i0 > XTOT - 1 ? XTOT - 1 : i0);
                int c1 = i1 < 0 ? 0 : (i1 > XTOT - 1 ? XTOT - 1 : i1);
                float t0 = x[c0];           // unconditional, address is safe
                float t1 = x[c1];
                bool v0 = y0 && (unsigned)(ixq0 + o) < W;
                bool v1 = y1 && (unsigned)(ixq1 + o) < W;
                bf.x = v0 ? t0 : 0.0f;      // pure v_cndmask, no exec games
                bf.y = v1 ? t1 : 0.0f;
            } else {
                bf.x = x[ib0 + o];          // interior: raw load, imm offset
                bf.y = x[ib1 + o];
            }
            acc[j] = __builtin_amdgcn_wmma_f32_16x16x4_f32(
                false, a, false, bf, (short)0, acc[j], false, false);
        }
    }
}

// ---------------------------------------------------------------------------
// Kernel 2: implicit-GEMM conv via V_WMMA_F32_16X16X4_F32.
//   M = 16 out-channels, K = 300 (75 x K=4 steps), N = 64 positions per wave.
//   Writes hardswish(conv + bias) to out, accumulates per-(b,oc) sums in ws.
// ---------------------------------------------------------------------------
__global__ __launch_bounds__(256) void conv_wmma_kernel(
    const float* __restrict__ x, const float* __restrict__ weight,
    const float* __restrict__ bias, float* __restrict__ out,
    float* __restrict__ ws)
{
    __shared__ float wlds[OUT_CH * KTOT];   // 19200 B
    __shared__ int2  kdec[KTOT];            // flat offset + packed (ky, kx-2)
    __shared__ float blds[OUT_CH];

    int tid = threadIdx.x;
    for (int i = tid; i < OUT_CH * KTOT; i += 256) wlds[i] = weight[i];
    for (int k = tid; k < KTOT; k += 256) {
        int c = k / 25, r = k % 25, ky = r / 5, kx = r % 5;
        kdec[k] = make_int2(c * HW + (ky - 2) * W + (kx - 2),
                            (ky << 8) | ((kx - 2) & 0xff));
    }
    if (tid < OUT_CH) blds[tid] = bias[tid];
    __syncthreads();

    int wave = tid >> 5, lane = tid & 31;
    int lh = lane >> 4;        // lane half: selects K pair and M offset
    int nl = lane & 15;

    // 18432 wave-tiles total: 576 tiles of 64 positions per image.
    // Force tile coordinates into SGPRs so base math stays on the SALU.
    int gtile = __builtin_amdgcn_readfirstlane(blockIdx.x * 8 + wave);
    int b   = gtile / 576;
    int t   = gtile - b * 576;
    int p0  = t * 64;
    int oh  = p0 / OW;
    int ow0 = p0 - oh * OW;    // 0, 64, or 128 — strip never crosses a row
    int oh2 = oh * 2;

    int posabs = b * (IN_CH * HW) + oh2 * W + 2 * ow0 + 2 * nl;
    int ixbase = 2 * ow0 + 2 * nl;

    v8f acc[4];
#pragma unroll
    for (int j = 0; j < 4; j++)
        acc[j] = (v8f){0.f, 0.f, 0.f, 0.f, 0.f, 0.f, 0.f, 0.f};

    bool yedge = (oh == 0) | (oh == OH - 1);
    if (yedge) {
        conv_loop<0, true>(x, wlds, kdec, nl, lh, oh2, posabs, ixbase, acc);
    } else if (ow0 == 0) {
        conv_loop<0, false>(x, wlds, kdec, nl, lh, oh2, posabs, ixbase, acc);
    } else if (ow0 == 64) {
        conv_loop<-1, false>(x, wlds, kdec, nl, lh, oh2, posabs, ixbase, acc);
    } else {
        conv_loop<3, false>(x, wlds, kdec, nl, lh, oh2, posabs, ixbase, acc);
    }

    // Epilogue: bias + hardswish, store, and per-channel sum for the mean.
    float csum[8];
#pragma unroll
    for (int r = 0; r < 8; r++) csum[r] = 0.0f;

    size_t ob = ((size_t)b * OUT_CH) * OHW + (size_t)oh * OW + ow0 + nl;
#pragma unroll
    for (int j = 0; j < 4; j++) {
#pragma unroll
        for (int r = 0; r < 8; r++) {
            int oc = r + lh * 8;
            float v  = acc[j][r] + blds[oc];
            float t6 = v + 3.0f;
            t6 = t6 < 0.0f ? 0.0f : (t6 > 6.0f ? 6.0f : t6);
            float h = v * t6 * (1.0f / 6.0f);
            out[ob + (size_t)oc * OHW + j * 16] = h;
            csum[r] += h;
        }
    }
#pragma unroll
    for (int r = 0; r < 8; r++) {
        float s = csum[r];
        s += __shfl_xor(s, 1, 32);
        s += __shfl_xor(s, 2, 32);
        s += __shfl_xor(s, 4, 32);
        s += __shfl_xor(s, 8, 32);   // reduced within each 16-lane half
        if (nl == 0)
            atomicAdd(&ws[WS_CONVSUM + b * OUT_CH + r + lh * 8], s);
    }
}

// ---------------------------------------------------------------------------
// Kernel 3: per-(b,oc) gating scale and residual-projection add term.
// ---------------------------------------------------------------------------
__global__ void combine_kernel(const float* __restrict__ res_w,
                               float* __restrict__ ws)
{
    int i = threadIdx.x;           // 512 threads = b*16 + oc
    int b = i >> 4, oc = i & 15;
    float rp = 0.0f;
    for (int c = 0; c < IN_CH; c++)
        rp += ws[WS_INSUM + b * IN_CH + c] * res_w[oc * IN_CH + c];
    rp *= (1.0f / (float)HW);

    float mean = ws[WS_CONVSUM + i] * (1.0f / (float)OHW);
    float g = mean < -0.5f ? -0.5f : (mean > 0.5f ? 0.5f : mean);
    ws[WS_SCALE + i] = 1.0f + g;
    ws[WS_ADD + i]   = rp;
}

// ---------------------------------------------------------------------------
// Kernel 4: out = clamp(act * (1+gating) + res_proj, -0.5, 0.5), float4 I/O.
// ---------------------------------------------------------------------------
__global__ __launch_bounds__(256) void finalize_kernel(
    float* __restrict__ out, const float* __restrict__ ws)
{
    size_t vi = (size_t)blockIdx.x * 256 + threadIdx.x;  // float4 index
    int plane = (int)(vi / (OHW / 4));                   // b*16 + oc
    float sc = ws[WS_SCALE + plane];
    float ad = ws[WS_ADD + plane];
    float4 v = ((const float4*)out)[vi];
    float o0 = v.x * sc + ad, o1 = v.y * sc + ad;
    float o2 = v.z * sc + ad, o3 = v.w * sc + ad;
    o0 = o0 < -0.5f ? -0.5f : (o0 > 0.5f ? 0.5f : o0);
    o1 = o1 < -0.5f ? -0.5f : (o1 > 0.5f ? 0.5f : o1);
    o2 = o2 < -0.5f ? -0.5f : (o2 > 0.5f ? 0.5f : o2);
    o3 = o3 < -0.5f ? -0.5f : (o3 > 0.5f ? 0.5f : o3);
    ((float4*)out)[vi] = make_float4(o0, o1, o2, o3);
}

// ---------------------------------------------------------------------------
extern "C" void kernel_launch(void* const* d_in, const int* in_sizes, int n_in,
                              void* d_out, int out_size, void* d_ws, size_t ws_size,
                              hipStream_t stream)
{
    const float* x      = (const float*)d_in[0];
    const float* weight = (const float*)d_in[1];
    const float* bias   = (const float*)d_in[2];
    const float* res_w  = (const float*)d_in[3];
    float* out = (float*)d_out;
    float* ws  = (float*)d_ws;

    hipMemsetAsync(ws, 0, WS_TOTAL * sizeof(float), stream);

    pool_input_kernel<<<BATCH * IN_CH, 256, 0, stream>>>(x, ws);

    // 18432 wave-tiles / 8 waves per block = 2304 blocks
    conv_wmma_kernel<<<2304, 256, 0, stream>>>(x, weight, bias, out, ws);

    combine_kernel<<<1, 512, 0, stream>>>(res_w, ws);

    // 32*16*36864 / 4 / 256 = 18432 blocks
    finalize_kernel<<<(BATCH * OUT_CH * OHW) / 4 / 256, 256, 0, stream>>>(out, ws);
}